// ReduceBoundingBoxes_53515292508437
// MI455X (gfx1250) — compile-verified
//
#include <hip/hip_runtime.h>
#include <stdint.h>

typedef float v2f __attribute__((ext_vector_type(2)));
typedef float v8f __attribute__((ext_vector_type(8)));

#define N_ENT 6400
#define N_PAD 8192
#define PROB_TH 0.9f
#define IOU_TH 0.5f

// ---------------------------------------------------------------------------
// Kernel 1: b5 = flat @ T^T via V_WMMA_F32_16X16X4_F32 (exact f32 FMA).
// A tile: 16 entries x 4 channels.  B: 4x16 coefficient matrix:
//   out0=f0, out1=f1, out2=f0+f2, out3=f1+f3 ; out4=f4 added as fixup.
// Exactly 400 waves (6400/16); launched as 50 blocks x 256 threads so EXEC
// is all-ones for every wave (WMMA requirement).
// ---------------------------------------------------------------------------
__global__ void b5_wmma_kernel(const float* __restrict__ x, float* __restrict__ b5) {
  const int lane = threadIdx.x & 31;
  const int wave = (int)((blockIdx.x * blockDim.x + threadIdx.x) >> 5);
  const int half = lane >> 4;     // 0: K={0,1}, 1: K={2,3}
  const int n    = lane & 15;     // A row (M) and B column (N)
  const int row  = (wave << 4) + n;
  const int k0   = half << 1;

  // A matrix 16x4 f32: lane l holds row M=l%16, channels K = 2*half, 2*half+1
  v2f a;
  a.x = x[k0 * N_ENT + row];
  a.y = x[(k0 + 1) * N_ENT + row];

  // B matrix 4x16 f32: VGPR v holds K = v + 2*half, column N = lane%16
  const float B0 = (n == 0 || n == 2) ? 1.0f : 0.0f;
  const float B1 = (n == 1 || n == 3) ? 1.0f : 0.0f;
  const float B2 = (n == 2) ? 1.0f : 0.0f;
  const float B3 = (n == 3) ? 1.0f : 0.0f;
  v2f b;
  b.x = half ? B2 : B0;
  b.y = half ? B3 : B1;

  v8f c = {};
  c = __builtin_amdgcn_wmma_f32_16x16x4_f32(false, a, false, b, (short)0, c,
                                            false, false);

  // D layout: VGPR v, lanes 0-15 -> M=v ; lanes 16-31 -> M=v+8 ; N = lane%16
  if (n < 5) {
#pragma unroll
    for (int v = 0; v < 8; ++v) {
      const int M = v + (half << 3);
      const int i = (wave << 4) + M;
      float val = c[v];
      if (n == 4) val += x[4 * N_ENT + i];   // out4 = f4 (K=4 fixup)
      b5[i * 5 + n] = val;
    }
  }
}

// ---------------------------------------------------------------------------
// Kernel 2: single workgroup (32 waves) — bitonic sort in LDS, greedy NMS by
// wave 0 (32-wide IoU probes with ballot_w32), parallel output write.
// ---------------------------------------------------------------------------
__device__ inline bool any32(bool p) {
  return __builtin_amdgcn_ballot_w32(p) != 0u;
}

__global__ __launch_bounds__(1024) void sort_nms_kernel(
    const float* __restrict__ x, const float* __restrict__ b5_all,
    float* __restrict__ b5s, float* __restrict__ area_s,
    float* __restrict__ out) {
  __shared__ unsigned long long keys[N_PAD];  // 64 KB of the WGP's 320 KB LDS
  const int tid = threadIdx.x;

  // --- Phase A: build sort keys: (score_bits << 16) | (0xFFFF - idx) -------
  for (int i = tid; i < N_PAD; i += 1024) {
    unsigned long long k = 0ull;
    if (i < N_ENT) {
      const float s = x[i];  // channel 0
      if (s > PROB_TH) {
        const unsigned int sb = __float_as_uint(s);  // positive: uint-sortable
        k = ((unsigned long long)sb << 16) |
            (unsigned long long)(0xFFFFu - (unsigned)i);
      }
    }
    keys[i] = k;
  }
  __syncthreads();

  // --- Phase B: bitonic sort, descending -----------------------------------
  for (int kk = 2; kk <= N_PAD; kk <<= 1) {
    for (int jj = kk >> 1; jj > 0; jj >>= 1) {
      for (int i = tid; i < N_PAD; i += 1024) {
        const int ixj = i ^ jj;
        if (ixj > i) {
          const unsigned long long a = keys[i];
          const unsigned long long b = keys[ixj];
          const bool up = (i & kk) == 0;
          if (up ? (a < b) : (a > b)) {
            keys[i]   = b;
            keys[ixj] = a;
          }
        }
      }
      __syncthreads();
    }
  }

  // --- Phase C: gather sorted b5 rows + areas into scratch -----------------
  for (int r = tid; r < N_ENT; r += 1024) {
    const unsigned long long k = keys[r];
    if (k != 0ull) {
      const int idx = 0xFFFF - (int)(k & 0xFFFFull);
      const float v0 = b5_all[idx * 5 + 0];
      const float v1 = b5_all[idx * 5 + 1];
      const float v2 = b5_all[idx * 5 + 2];
      const float v3 = b5_all[idx * 5 + 3];
      const float v4 = b5_all[idx * 5 + 4];
      b5s[r * 5 + 0] = v0;
      b5s[r * 5 + 1] = v1;
      b5s[r * 5 + 2] = v2;
      b5s[r * 5 + 3] = v3;
      b5s[r * 5 + 4] = v4;
      // boxes = b5[1:5] treated as [x1,y1,x2,y2]
      area_s[r] = fmaxf(v3 - v1, 0.0f) * fmaxf(v4 - v2, 0.0f);
    } else {
      area_s[r] = 0.0f;
    }
  }
  __threadfence_block();
  __syncthreads();

  // --- Phase D: greedy NMS, wave 0 only ------------------------------------
  if (tid < 32) {
    // M = number of valid (nonzero-key) entries; keys are sorted descending.
    int lo = 0, hi = N_PAD;
    while (lo < hi) {
      const int mid = (lo + hi) >> 1;
      if (keys[mid] != 0ull) lo = mid + 1; else hi = mid;
    }
    const int M = lo;       // <= 6400
    const int lane = tid;

    for (int i = 0; i < M; ++i) {
      const float bi0 = b5s[i * 5 + 1];
      const float bi1 = b5s[i * 5 + 2];
      const float bi2 = b5s[i * 5 + 3];
      const float bi3 = b5s[i * 5 + 4];
      const float ai  = area_s[i];
      bool suppressed = false;
      for (int jb = 0; jb < i && !suppressed; jb += 32) {
        const int j = jb + lane;
        bool pred = false;
        if (j < i && (keys[j] >> 63)) {     // already-kept box
          const float bj0 = b5s[j * 5 + 1];
          const float bj1 = b5s[j * 5 + 2];
          const float bj2 = b5s[j * 5 + 3];
          const float bj3 = b5s[j * 5 + 4];
          const float ltx = fmaxf(bi0, bj0);
          const float lty = fmaxf(bi1, bj1);
          const float rbx = fminf(bi2, bj2);
          const float rby = fminf(bi3, bj3);
          const float w = fmaxf(rbx - ltx, 0.0f);
          const float h = fmaxf(rby - lty, 0.0f);
          const float inter = w * h;
          const float uni = ai + area_s[j] - inter;
          const float iou = inter / fmaxf(uni, 1e-9f);
          pred = iou > IOU_TH;
        }
        if (any32(pred)) suppressed = true;
      }
      if (!suppressed && lane == 0) {
        keys[i] |= 0x8000000000000000ull;   // keep bit
      }
      asm volatile("s_wait_dscnt 0" ::: "memory");
    }
  }
  __syncthreads();

  // --- Phase E: write output (kept rows = sorted b5, else zeros) -----------
  for (int r = tid; r < N_ENT; r += 1024) {
    const bool kept = (keys[r] >> 63) != 0ull;
#pragma unroll
    for (int c = 0; c < 5; ++c) {
      out[r * 5 + c] = kept ? b5s[r * 5 + c] : 0.0f;
    }
  }
}

extern "C" void kernel_launch(void* const* d_in, const int* in_sizes, int n_in,
                              void* d_out, int out_size, void* d_ws, size_t ws_size,
                              hipStream_t stream) {
  (void)in_sizes; (void)n_in; (void)out_size; (void)ws_size;
  const float* x = (const float*)d_in[0];
  float* out = (float*)d_out;
  float* ws = (float*)d_ws;
  float* b5_all = ws;              // 32000 floats
  float* b5s    = ws + 32000;      // 32000 floats (sorted rows)
  float* area_s = ws + 64000;      // 6400 floats

  // 50 blocks x 256 threads = exactly 400 waves = 6400/16 WMMA tiles.
  b5_wmma_kernel<<<50, 256, 0, stream>>>(x, b5_all);
  sort_nms_kernel<<<1, 1024, 0, stream>>>(x, b5_all, b5s, area_s, out);
}